// LNODEfunc3_WithInput_84473416777805
// MI455X (gfx1250) — compile-verified
//
#include <hip/hip_runtime.h>
#include <math.h>

typedef float v2f __attribute__((ext_vector_type(2)));
typedef float v8f __attribute__((ext_vector_type(8)));

// workspace float-slot layout
#define WS_M1   0
#define WS_M2   9
#define WS_M3   18
#define WS_M4   27
#define WS_A1   36
#define WS_B1   37
#define WS_A2   38
#define WS_B2   39
#define WS_FRAC 40
#define WS_II   41   // int bits
#define WS_UN   42   // int bits

// ---------------------------------------------------------------------------
// One-wave precompute: M1..M4 (3x3), A1,B1,A2,B2, interpolation index/frac.
// ~1k FLOPs total; negligible.
// ---------------------------------------------------------------------------
__global__ void lnode_precompute(const float* __restrict__ t, const int* __restrict__ Dt,
                                 const float* __restrict__ R, const float* __restrict__ m,
                                 const float* __restrict__ W1, const float* __restrict__ W2,
                                 const float* __restrict__ W3, const float* __restrict__ W4,
                                 const float* __restrict__ D1, const float* __restrict__ D2,
                                 const float* __restrict__ D3, const float* __restrict__ D4,
                                 const float* __restrict__ Wb1, const float* __restrict__ Wb2,
                                 const float* __restrict__ Wout,
                                 float* __restrict__ ws, int T) {
  if (threadIdx.x != 0 || blockIdx.x != 0) return;
  const int F = 20;
  const float EPS = 1e-6f;

  // P = R @ m : [3(k), 20(f)]  (base[f] vector = P[:,f])
  float P[3][20];
  for (int k = 0; k < 3; ++k)
    for (int f = 0; f < F; ++f) {
      float acc = 0.f;
      for (int j = 0; j < 3; ++j) acc += R[k*3 + j] * m[j*F + f];
      P[k][f] = acc;
    }

  const float* Ws[4] = {W1, W2, W3, W4};
  const float* Ds[4] = {D1, D2, D3, D4};
  const int    Mo[4] = {WS_M1, WS_M2, WS_M3, WS_M4};

  for (int i = 0; i < 4; ++i) {
    // Y[o][k] = sum_f W[o,f] * P[k][f]   (ln_linear on base)
    float Y[3][3], Dd[3][3], res[3][3];
    for (int o = 0; o < 3; ++o)
      for (int k = 0; k < 3; ++k) {
        float acc = 0.f;
        for (int f = 0; f < F; ++f) acc += Ws[i][o*F + f] * P[k][f];
        Y[o][k] = acc;
      }
    // Dd[g][k] = sum_o D[g,o] * Y[o][k]
    for (int g = 0; g < 3; ++g)
      for (int k = 0; k < 3; ++k) {
        float acc = 0.f;
        for (int o = 0; o < 3; ++o) acc += Ds[i][g*3 + o] * Y[o][k];
        Dd[g][k] = acc;
      }
    // Killing-ReLU per channel c
    for (int c = 0; c < 3; ++c) {
      float kxd = 0.f, kdd = 0.f;
      for (int k = 0; k < 3; ++k) { kxd += Y[c][k]*Dd[c][k]; kdd += Dd[c][k]*Dd[c][k]; }
      kxd *= -2.f; kdd *= -2.f;
      if (kxd < 0.f) {
        for (int k = 0; k < 3; ++k) res[c][k] = Y[c][k];
      } else {
        float s = kxd / (kdd - EPS);
        for (int k = 0; k < 3; ++k) res[c][k] = Y[c][k] - s * Dd[c][k];
      }
    }
    // m_i[k][c] = res[c][k];  M_i = m_i m_i^T  ->  M[a][b] = sum_c res[c][a]*res[c][b]
    float* M = ws + Mo[i];
    for (int a = 0; a < 3; ++a)
      for (int b = 0; b < 3; ++b) {
        float acc = 0.f;
        for (int c = 0; c < 3; ++c) acc += res[c][a] * res[c][b];
        M[a*3 + b] = acc;
      }
  }

  float A1 = 0.f, B1 = 0.f, A2 = 0.f, B2 = 0.f;
  for (int f = 0; f < F; ++f) {
    float wo = Wout[f];
    A1 += wo * Wb1[f] * Wb1[f];
    B1 += wo * Wb1[f];
    A2 += wo * Wb2[f] * Wb2[f];
    B2 += wo * Wb2[f];
  }
  ws[WS_A1] = A1; ws[WS_B1] = B1; ws[WS_A2] = A2; ws[WS_B2] = B2;

  // interpolate_u index/frac (faithful to reference)
  float ts  = t[0];
  float Dtf = (float)Dt[0];
  int ii = (int)floorf(ts / Dtf);
  ii = ii < 0 ? 0 : (ii > T - 1 ? T - 1 : ii);
  int un = (ii + 1 < T - 1) ? ii + 1 : T - 1;
  float frac = (ts - (float)ii * Dtf) / Dtf;
  if (ii >= T - 1) frac = 0.f;   // jnp.where(ii >= T-1, u_i, ...)
  ws[WS_FRAC] = frac;
  ((int*)ws)[WS_II] = ii;
  ((int*)ws)[WS_UN] = un;
}

// ---------------------------------------------------------------------------
// Streaming kernel: 16 points per wave.
// WMMA #1: A=[M1;M2] (16x4, rows 0..5 used)  x  B=[x-vectors] (4x16)
// WMMA #2: A=[M3;M4]                         x  B=[lerped u-vectors]
// D rows 0..2 -> Ma*v, rows 3..5 -> Mb*v, per column (point).
// Then cross products + linear combination in VALU, store 3 floats/point.
// ---------------------------------------------------------------------------
__global__ void lnode_main(const float* __restrict__ x, const float* __restrict__ u,
                           const float* __restrict__ ws, float* __restrict__ out, int B) {
  const int lane = threadIdx.x & 31;
  const int wave = threadIdx.x >> 5;
  const int n    = lane & 15;
  const bool hi  = lane >= 16;

  long p = ((long)blockIdx.x * (blockDim.x >> 5) + wave) * 16 + n;
  const bool valid = p < (long)B;
  const long pc = valid ? p : (long)B - 1;   // clamp so every lane loads safely

  // uniform scalars (scalar loads)
  const float frac = ws[WS_FRAC];
  const int   ii   = ((const int*)ws)[WS_II];
  const int   un   = ((const int*)ws)[WS_UN];
  const float A1 = ws[WS_A1], B1 = ws[WS_B1], A2 = ws[WS_A2], B2 = ws[WS_B2];

  // ---- A operands: 16x4 f32 layout: lane = M-row, lane-half selects K {0,1}/{2,3}
  const int r  = n;             // M row
  const int k0 = hi ? 2 : 0;    // K base for this lane half
  v2f Ax = {0.f, 0.f};
  v2f Au = {0.f, 0.f};
  if (r < 6) {
    const float* mrx = (r < 3) ? (ws + WS_M1 + r*3) : (ws + WS_M2 + (r-3)*3);
    const float* mru = (r < 3) ? (ws + WS_M3 + r*3) : (ws + WS_M4 + (r-3)*3);
    Ax.x = mrx[k0];             // k0 in {0,2} -> real column
    Au.x = mru[k0];
    if (!hi) { Ax.y = mrx[1]; Au.y = mru[1]; }   // K=3 (hi half .y) stays 0 (pad)
  }

  // ---- B operands: 4x16 f32 layout: lane = N-column, lane-half selects K {0,1}/{2,3}
  const float* xp  = x + pc * 3;
  const float* uip = u + ((long)ii * B + pc) * 3;
  const float* unp = u + ((long)un * B + pc) * 3;

  float xv0 = 0.f, xv1 = 0.f, xv2 = 0.f;
  float uv0 = 0.f, uv1 = 0.f, uv2 = 0.f;
  v2f Bx, Bu;
  if (!hi) {
    xv0 = xp[0]; xv1 = xp[1]; xv2 = xp[2];
    float a0 = uip[0], a1 = uip[1], a2 = uip[2];
    float b0 = unp[0], b1 = unp[1], b2 = unp[2];
    uv0 = a0 + frac * (b0 - a0);
    uv1 = a1 + frac * (b1 - a1);
    uv2 = a2 + frac * (b2 - a2);
    Bx.x = xv0; Bx.y = xv1;     // K=0,1
    Bu.x = uv0; Bu.y = uv1;
  } else {
    float xz = xp[2];
    float az = uip[2], bz = unp[2];
    Bx.x = xz;                  // K=2
    Bx.y = 0.f;                 // K=3 pad
    Bu.x = az + frac * (bz - az);
    Bu.y = 0.f;
  }

  // ---- WMMA (EXEC is all-1s here; divergent paths above have reconverged)
  v8f cz = {};
  v8f dX = __builtin_amdgcn_wmma_f32_16x16x4_f32(
      /*neg_a=*/false, Ax, /*neg_b=*/false, Bx, /*c_mod=*/(short)0, cz,
      /*reuse_a=*/false, /*reuse_b=*/false);
  v8f dU = __builtin_amdgcn_wmma_f32_16x16x4_f32(
      false, Au, false, Bu, (short)0, cz, false, false);

  // lanes 0..15 hold D rows 0..7 at column n: rows 0..2 = Ma*v, rows 3..5 = Mb*v
  if (!hi && valid) {
    float v1x = dX[0], v1y = dX[1], v1z = dX[2];
    float v2x = dX[3], v2y = dX[4], v2z = dX[5];
    float cxx = v1y * v2z - v1z * v2y;
    float cxy = v1z * v2x - v1x * v2z;
    float cxz = v1x * v2y - v1y * v2x;

    float w1x = dU[0], w1y = dU[1], w1z = dU[2];
    float w2x = dU[3], w2y = dU[4], w2z = dU[5];
    float cux = w1y * w2z - w1z * w2y;
    float cuy = w1z * w2x - w1x * w2z;
    float cuz = w1x * w2y - w1y * w2x;

    float* op = out + p * 3;
    op[0] = A1 * cxx + B1 * xv0 + A2 * cux + B2 * uv0;
    op[1] = A1 * cxy + B1 * xv1 + A2 * cuy + B2 * uv1;
    op[2] = A1 * cxz + B1 * xv2 + A2 * cuz + B2 * uv2;
  }
}

// ---------------------------------------------------------------------------
extern "C" void kernel_launch(void* const* d_in, const int* in_sizes, int n_in,
                              void* d_out, int out_size, void* d_ws, size_t ws_size,
                              hipStream_t stream) {
  const float* t    = (const float*)d_in[0];
  const float* x    = (const float*)d_in[1];
  const float* u    = (const float*)d_in[2];
  const int*   Dt   = (const int*)  d_in[3];
  const float* R    = (const float*)d_in[4];
  const float* m    = (const float*)d_in[5];
  const float* W1   = (const float*)d_in[6];
  const float* W2   = (const float*)d_in[7];
  const float* W3   = (const float*)d_in[8];
  const float* W4   = (const float*)d_in[9];
  const float* D1   = (const float*)d_in[10];
  const float* D2   = (const float*)d_in[11];
  const float* D3   = (const float*)d_in[12];
  const float* D4   = (const float*)d_in[13];
  const float* Wb1  = (const float*)d_in[14];
  const float* Wb2  = (const float*)d_in[15];
  const float* Wout = (const float*)d_in[16];

  const int B = in_sizes[1] / 3;                 // x: [B,1,3]
  const int T = in_sizes[2] / in_sizes[1];       // u: [T,B,1,3]
  float* ws = (float*)d_ws;
  float* out = (float*)d_out;

  lnode_precompute<<<1, 32, 0, stream>>>(t, Dt, R, m, W1, W2, W3, W4,
                                         D1, D2, D3, D4, Wb1, Wb2, Wout, ws, T);

  const int threads = 256;                       // 8 waves * 16 points = 128 pts/block
  const int ptsPerBlock = (threads / 32) * 16;
  const int grid = (B + ptsPerBlock - 1) / ptsPerBlock;
  lnode_main<<<grid, threads, 0, stream>>>(x, u, ws, out, B);
}